// MyGCNConv_50912542327337
// MI455X (gfx1250) — compile-verified
//
#include <hip/hip_runtime.h>

typedef __attribute__((ext_vector_type(2))) float v2f;
typedef __attribute__((ext_vector_type(8))) float v8f;

#define IN_DIM 128
#define OUT_DIM 128

// ---------------------------------------------------------------- degree init
__global__ void k_init_deg(float* __restrict__ deg, int n) {
    int i = blockIdx.x * blockDim.x + threadIdx.x;
    if (i < n) deg[i] = 1.0f;               // self loop contributes 1 to deg
}

// ------------------------------------------------------- histogram of sources
__global__ void k_count(const int* __restrict__ src, float* __restrict__ deg, int E) {
    int e = blockIdx.x * blockDim.x + threadIdx.x;
    if (e < E) atomicAdd(&deg[src[e]], 1.0f);
}

// ------------------------------------------------------------ dis = rsqrt(deg)
__global__ void k_rsqrt(float* __restrict__ deg, int n) {
    int i = blockIdx.x * blockDim.x + threadIdx.x;
    if (i < n) deg[i] = rsqrtf(deg[i]);     // deg >= 1 always (self loops)
}

// ------------------- pair-interleaved transpose: Wtp[k/2][o] = {W[o][k], W[o][k+1]}
// Element layout (floats): Wtp[(k>>1)*256 + o*2 + (k&1)] = W[o*128 + k]
// so a B fragment for even k0 is ONE contiguous float2 at (k0/2)*128 + col.
__global__ void k_transpose(const float* __restrict__ W, float* __restrict__ Wtp) {
    int idx = blockIdx.x * blockDim.x + threadIdx.x;   // 16384 threads
    int o = idx >> 7;
    int k = idx & 127;
    Wtp[(k >> 1) * (2 * OUT_DIM) + o * 2 + (k & 1)] = W[o * IN_DIM + k];
}

// ------------------------------------------------------------- WMMA f32 GEMM
// h[N,128] = x[N,128] @ W^T; one wave per 16-row tile, 8 N-tiles per wave.
__global__ __launch_bounds__(256) void k_gemm(const float* __restrict__ x,
                                              const float* __restrict__ Wtp,
                                              float* __restrict__ h,
                                              int nTiles) {
    __shared__ float sW[IN_DIM * OUT_DIM];             // 64 KB of 320 KB WGP LDS

    // cooperative stage of pair-interleaved W into LDS (float4, coalesced)
    {
        const float4* g4 = (const float4*)Wtp;
        float4* s4 = (float4*)sW;
        for (int i = threadIdx.x; i < (IN_DIM * OUT_DIM) / 4; i += blockDim.x)
            s4[i] = g4[i];
    }
    __syncthreads();

    const int wid   = threadIdx.x >> 5;
    const int lane  = threadIdx.x & 31;
    const int tileM = blockIdx.x * 8 + wid;
    if (tileM >= nTiles) return;                       // uniform per wave: EXEC stays all-1s

    const int l16   = lane & 15;
    const int halfK = lane >> 4;                       // A layout: lanes16-31 hold K+2
    const float* xrow = x + (size_t)(tileM * 16 + l16) * IN_DIM;
    const v2f* sB = (const v2f*)sW;                    // [64][128] float2 pairs

    v8f acc[8];
#pragma unroll
    for (int t = 0; t < 8; ++t)
        acc[t] = (v8f){0.f, 0.f, 0.f, 0.f, 0.f, 0.f, 0.f, 0.f};

    for (int kb = 0; kb < IN_DIM; kb += 4) {
        const int k0 = kb + halfK * 2;                 // even -> 8B aligned
        float2 av = *(const float2*)(xrow + k0);       // A frag: (M=l16, K=k0..k0+1)
        v2f a; a.x = av.x; a.y = av.y;
        const v2f* bRow = sB + (k0 >> 1) * OUT_DIM + l16;
#pragma unroll
        for (int t = 0; t < 8; ++t) {
            v2f b = bRow[t * 16];                      // single ds_load_b64
            acc[t] = __builtin_amdgcn_wmma_f32_16x16x4_f32(
                /*neg_a=*/false, a, /*neg_b=*/false, b,
                /*c_mod=*/(short)0, acc[t], /*reuse_a=*/false, /*reuse_b=*/false);
        }
    }

    // C/D layout: VGPR v -> row (v + 8*halfK), col = lane%16
    const int rowBase = tileM * 16 + halfK * 8;
#pragma unroll
    for (int t = 0; t < 8; ++t) {
        float* hp = h + (size_t)rowBase * OUT_DIM + t * 16 + l16;
#pragma unroll
        for (int v = 0; v < 8; ++v)
            hp[(size_t)v * OUT_DIM] = acc[t][v];
    }
}

// -------------------------------------- scalar tail GEMM for rows >= nTiles*16
__global__ void k_gemm_tail(const float* __restrict__ x, const float* __restrict__ Wtp,
                            float* __restrict__ h, int rowStart, int n) {
    int idx = blockIdx.x * blockDim.x + threadIdx.x;
    int nrows = n - rowStart;
    if (idx >= nrows * OUT_DIM) return;
    int r = rowStart + idx / OUT_DIM;
    int o = idx % OUT_DIM;
    float s = 0.f;
    for (int k = 0; k < IN_DIM; ++k)
        s += x[(size_t)r * IN_DIM + k] * Wtp[(k >> 1) * (2 * OUT_DIM) + o * 2 + (k & 1)];
    h[(size_t)r * OUT_DIM + o] = s;
}

// --------------------------- out[i] = h[i] * dis[i]^2  (self-loop term + init)
__global__ void k_selfloop(const float* __restrict__ h, const float* __restrict__ dis,
                           float* __restrict__ out, int n) {
    int idx = blockIdx.x * blockDim.x + threadIdx.x;   // n*32 float4 slots
    if (idx >= n * 32) return;
    int i = idx >> 5;
    float d = dis[i];
    float norm = d * d;
    float4 v = ((const float4*)h)[idx];
    ((float4*)out)[idx] = make_float4(v.x * norm, v.y * norm, v.z * norm, v.w * norm);
}

// ------------------------ per-edge gather/scale/scatter-add: one wave per edge
__global__ __launch_bounds__(256) void k_scatter(const int* __restrict__ src,
                                                 const int* __restrict__ tgt,
                                                 const float* __restrict__ h,
                                                 const float* __restrict__ dis,
                                                 float* __restrict__ out, int E) {
    int lane = threadIdx.x & 31;
    int e = blockIdx.x * (blockDim.x >> 5) + (threadIdx.x >> 5);
    if (e >= E) return;
    int s = src[e];
    int t = tgt[e];
    float norm = dis[s] * dis[t];
    float4 v = ((const float4*)(h + (size_t)s * OUT_DIM))[lane];
    float* ob = out + (size_t)t * OUT_DIM + lane * 4;
    atomicAdd(ob + 0, v.x * norm);
    atomicAdd(ob + 1, v.y * norm);
    atomicAdd(ob + 2, v.z * norm);
    atomicAdd(ob + 3, v.w * norm);
}

extern "C" void kernel_launch(void* const* d_in, const int* in_sizes, int n_in,
                              void* d_out, int out_size, void* d_ws, size_t ws_size,
                              hipStream_t stream) {
    const float* x  = (const float*)d_in[0];
    const int*   ei = (const int*)d_in[1];
    const float* W  = (const float*)d_in[2];
    float* out = (float*)d_out;

    const int n = in_sizes[0] / IN_DIM;
    const int E = in_sizes[1] / 2;
    const int* src = ei;
    const int* tgt = ei + E;

    // workspace: h [n*128] | dis [n, padded to 4] | Wtp [128*128]
    float* h   = (float*)d_ws;
    size_t disPad = ((size_t)n + 3) & ~(size_t)3;
    float* dis = h + (size_t)n * OUT_DIM;
    float* Wtp = dis + disPad;

    k_init_deg <<<(n + 255) / 256, 256, 0, stream>>>(dis, n);
    k_count    <<<(E + 255) / 256, 256, 0, stream>>>(src, dis, E);
    k_rsqrt    <<<(n + 255) / 256, 256, 0, stream>>>(dis, n);
    k_transpose<<<(IN_DIM * OUT_DIM) / 256, 256, 0, stream>>>(W, Wtp);

    const int nTiles = n / 16;
    if (nTiles > 0)
        k_gemm<<<(nTiles + 7) / 8, 256, 0, stream>>>(x, Wtp, h, nTiles);
    const int tail = n - nTiles * 16;
    if (tail > 0)
        k_gemm_tail<<<(tail * OUT_DIM + 255) / 256, 256, 0, stream>>>(x, Wtp, h, nTiles * 16, n);

    k_selfloop<<<((size_t)n * 32 + 255) / 256, 256, 0, stream>>>(h, dis, out, n);
    k_scatter <<<(E + 7) / 8, 256, 0, stream>>>(src, tgt, h, dis, out, E);
}